// Encoder_31421980738061
// MI455X (gfx1250) — compile-verified
//
#include <hip/hip_runtime.h>

// -------------------------------------------------------------------------
// Encoder for B=16, N=128, H=128.
//
// Output layout in d_out (flat, reference return order):
//   node_hidden  [16,128,128]      @ 0         (262144 f32)
//   edge_hidden  [16,128,128,128]  @ 262144    (33554432 f32)  <- 128 MiB, dominates
//   graph_hidden [16,128]          @ 33816576  (2048 f32)
//   adj_mat      [16,128,128]      @ 33818624  (262144 f32)
//
// Perf model (MI455X, 23.3 TB/s HBM): ~130 MiB stores + ~3 MiB loads
// => ~5.8 us, pure store-bandwidth bound. Kernel 1 streams edge_hidden with
// one wave per 512B output row (32 lanes x float4 = global_store_b128) and
// fuses adj_mat. Kernel 2 computes node_hidden; the W_pi[pi] gather runs as
// one_hot(pi) @ W_pi on the f32 WMMA pipe (bit-exact: products are 1*w or 0),
// with W_pi staged in LDS in a K-pair-interleaved layout so every WMMA's B
// operand is a single ds_load_b64 with an immediate offset (no address VALU).
// -------------------------------------------------------------------------

typedef __attribute__((ext_vector_type(2))) float v2f;
typedef __attribute__((ext_vector_type(8))) float v8f;

#define BB 16
#define NN 128
#define HH 128

// ---------------- Kernel 1: edge_hidden + adj_mat (store-BW bound) --------
__global__ __launch_bounds__(256) void edge_adj_kernel(
    const float* __restrict__ A, const int* __restrict__ adj,
    const int* __restrict__ pi,
    const float* __restrict__ W_A, const float* __restrict__ b_A,
    const float* __restrict__ W_adj, const float* __restrict__ b_adj,
    float* __restrict__ edge_out, float* __restrict__ adj_out)
{
    const int lane = threadIdx.x & 31;
    const int wave = threadIdx.x >> 5;
    const int wgid = blockIdx.x * 8 + wave;   // global wave id

    // Prefetch the strided (transposed) adj reads for this wave's 4 edges;
    // the streaming A/adj reads need no help.
    if (lane == 0) {
#pragma unroll
        for (int t = 0; t < 4; ++t) {
            const int e = wgid * 4 + t;
            const int b = e >> 14;
            const int i = (e >> 7) & 127;
            const int j = e & 127;
            __builtin_prefetch(&adj[(b * NN + j) * NN + i], 0, 0); // global_prefetch_b8
        }
    }

    // Per-lane weight/bias registers: lane l owns h = 4l..4l+3 (L2-resident).
    const float4 wA = ((const float4*)W_A)[lane];
    const float4 wJ = ((const float4*)W_adj)[lane];
    float4 bs;
    {
        const float4 ba = ((const float4*)b_A)[lane];
        const float4 bj = ((const float4*)b_adj)[lane];
        bs.x = ba.x + bj.x; bs.y = ba.y + bj.y;
        bs.z = ba.z + bj.z; bs.w = ba.w + bj.w;
    }

    // 4 consecutive edges per wave: 262144 edges / (8 waves * 4) = 8192 blocks.
#pragma unroll
    for (int t = 0; t < 4; ++t) {
        const int e = wgid * 4 + t;           // == (b*128 + i)*128 + j
        const int b = e >> 14;
        const int i = (e >> 7) & 127;
        const int j = e & 127;

        const float a   = A[e];
        const int   ad  = adj[e];
        const float adf = (float)ad;

        float4 o;
        o.x = fmaf(a, wA.x, fmaf(adf, wJ.x, bs.x));
        o.y = fmaf(a, wA.y, fmaf(adf, wJ.y, bs.y));
        o.z = fmaf(a, wA.z, fmaf(adf, wJ.z, bs.z));
        o.w = fmaf(a, wA.w, fmaf(adf, wJ.w, bs.w));
        ((float4*)edge_out)[e * 32 + lane] = o;   // 512B row, b128 stores

        if (lane == 0) {
            const int adT = adj[(b * NN + j) * NN + i];
            const int pii = pi[b * NN + i];
            const int pij = pi[b * NN + j];
            const bool on = (i == j) | (pii == j) | (pij == i) | (ad != 0) | (adT != 0);
            adj_out[e] = on ? 1.0f : 0.0f;
        }
    }
}

// ------- Kernel 2: node_hidden via one_hot(pi) @ W_pi on WMMA + graph -----
__global__ __launch_bounds__(256) void node_graph_kernel(
    const float* __restrict__ pos, const float* __restrict__ s,
    const int* __restrict__ pi, const float* __restrict__ key_g,
    const float* __restrict__ W_pos, const float* __restrict__ b_pos,
    const float* __restrict__ W_s, const float* __restrict__ b_s,
    const float* __restrict__ W_pi, const float* __restrict__ b_pi,
    const float* __restrict__ W_g, const float* __restrict__ b_g,
    float* __restrict__ node_out, float* __restrict__ graph_out)
{
    // W_pi staged K-pair-interleaved: lds_w[p*256 + 2n + (k&1)] = W_pi[k][n],
    // p = k>>1. A lane's B operand (rows kb,kb+1 at column n) is then one
    // contiguous 8-byte LDS read at byte offset p*1024 + 8n -> ds_load_b64
    // with immediate offset (k0 varies only the immediate: 0..63488).
    __shared__ __align__(16) float s_wpi[NN * HH];   // 64 KiB
    __shared__ int   s_pi[NN];
    __shared__ float s_pos[NN];
    __shared__ float s_s[NN];

    const int b   = blockIdx.x;   // one block per batch
    const int tid = threadIdx.x;

    if (tid < NN) {
        s_pi[tid]  = pi[b * NN + tid];
        s_pos[tid] = pos[b * NN + tid];
        s_s[tid]   = s[b * NN + tid];
        // graph_hidden: trivial, piggyback here
        graph_out[b * HH + tid] = fmaf(key_g[b], W_g[tid], b_g[tid]);
    }

    // Stage + interleave W_pi: 2048 tasks (64 row-pairs x 32 column-quads),
    // 8 per thread; each task: 2x global_load_b128 + 2x ds_store_b128.
#pragma unroll
    for (int it = 0; it < 8; ++it) {
        const int id = tid + it * 256;      // 0..2047
        const int p  = id >> 5;             // row pair 0..63
        const int q  = id & 31;             // column quad 0..31
        const float4 ra = ((const float4*)(W_pi + (2 * p)     * HH))[q];
        const float4 rb = ((const float4*)(W_pi + (2 * p + 1) * HH))[q];
        float4 lo, hi;
        lo.x = ra.x; lo.y = rb.x; lo.z = ra.y; lo.w = rb.y;
        hi.x = ra.z; hi.y = rb.z; hi.z = ra.w; hi.w = rb.w;
        ((float4*)s_wpi)[p * 64 + 2 * q]     = lo;
        ((float4*)s_wpi)[p * 64 + 2 * q + 1] = hi;
    }
    __syncthreads();

    const int wave = tid >> 5;        // 8 waves -> 8 M-tile rows
    const int lane = tid & 31;
    const int half = lane >> 4;       // upper/lower half-wave
    const int lp   = lane & 15;
    const int m0   = wave * 16;

    const int pim = s_pi[m0 + lp];    // A-operand row (M = m0+lp) pointer value

    for (int t = 0; t < 8; ++t) {     // 8 N-tiles of 16
        const int n0 = t * 16;
        const int n  = n0 + lp;
        v8f c = {};

        // Fixed LDS base for this tile; K loop varies only the imm offset.
        const float* bbase = &s_wpi[half * 256 + 2 * n];

        // A 16x4 layout: VGPR v, half h -> K = k0 + h*2 + v ; M = lp.
        // B 4x16 layout: VGPR v, half h -> K = k0 + h*2 + v ; N = n0 + lp.
#pragma unroll
        for (int k0 = 0; k0 < NN; k0 += 4) {
            const int kb = k0 + half * 2;
            v2f a;
            a.x = (pim == kb)     ? 1.0f : 0.0f;
            a.y = (pim == kb + 1) ? 1.0f : 0.0f;
            const v2f bm = *(const v2f*)(bbase + (k0 >> 1) * 256); // ds_load_b64
            c = __builtin_amdgcn_wmma_f32_16x16x4_f32(
                    /*neg_a=*/false, a, /*neg_b=*/false, bm,
                    /*c_mod=*/(short)0, c, /*reuse_a=*/false, /*reuse_b=*/false);
        }

        // Add rank-1 terms + fused bias, then store.
        const float wpn = W_pos[n];
        const float wsn = W_s[n];
        const float bn  = b_pos[n] + b_s[n] + b_pi[n];
#pragma unroll
        for (int r = 0; r < 8; ++r) {
            const int m = m0 + r + half * 8;   // C/D layout: M = r (+8 upper half)
            const float v = c[r] + fmaf(s_pos[m], wpn, fmaf(s_s[m], wsn, bn));
            node_out[(b * NN + m) * NN + n] = v;
        }
    }
}

// --------------------------------- launch ---------------------------------
extern "C" void kernel_launch(void* const* d_in, const int* in_sizes, int n_in,
                              void* d_out, int out_size, void* d_ws, size_t ws_size,
                              hipStream_t stream) {
    (void)in_sizes; (void)n_in; (void)d_ws; (void)ws_size; (void)out_size;

    const float* pos   = (const float*)d_in[0];
    const float* s     = (const float*)d_in[1];
    const float* A     = (const float*)d_in[2];
    const int*   adj   = (const int*)  d_in[3];
    const int*   pi    = (const int*)  d_in[4];
    const float* key_g = (const float*)d_in[5];
    const float* W_pos = (const float*)d_in[6];
    const float* b_pos = (const float*)d_in[7];
    const float* W_s   = (const float*)d_in[8];
    const float* b_s   = (const float*)d_in[9];
    const float* W_A   = (const float*)d_in[10];
    const float* b_A   = (const float*)d_in[11];
    const float* W_adj = (const float*)d_in[12];
    const float* b_adj = (const float*)d_in[13];
    const float* W_pi  = (const float*)d_in[14];
    const float* b_pi  = (const float*)d_in[15];
    const float* W_g   = (const float*)d_in[16];
    const float* b_g   = (const float*)d_in[17];

    float* out        = (float*)d_out;
    float* node_out   = out;                        // 262144
    float* edge_out   = out + 262144;               // 33554432
    float* graph_out  = out + 262144 + 33554432;    // 2048
    float* adjmat_out = graph_out + 2048;           // 262144

    // 16*128*128 edges / (8 waves * 4 edges) = 8192 blocks
    edge_adj_kernel<<<8192, 256, 0, stream>>>(
        A, adj, pi, W_A, b_A, W_adj, b_adj, edge_out, adjmat_out);

    node_graph_kernel<<<BB, 256, 0, stream>>>(
        pos, s, pi, key_g, W_pos, b_pos, W_s, b_s, W_pi, b_pi, W_g, b_g,
        node_out, graph_out);
}